// NormStability_41738492182535
// MI455X (gfx1250) — compile-verified
//
#include <hip/hip_runtime.h>
#include <hip/hip_bf16.h>

// Pure streaming elementwise kernel: 201 MB of traffic, ~8.6 us floor at
// 23.3 TB/s HBM. No matmul -> WMMA not applicable; the gfx1250-optimal path
// is wave32 + 128-bit global loads/stores with non-temporal (NT) hints so the
// single-use stream does not pollute WGP$/L2.

typedef float v4f __attribute__((ext_vector_type(4)));

__device__ __forceinline__ float norm_stab_elem(float v, float s, int E,
                                                float Lf, float inv_sl) {
    // p window (bipolar)
    float p    = (v + 1.0f) * 0.5f;
    float minp = fmaxf(p - 0.025f, 0.0f);
    float maxp = fminf(p + 0.025f, 1.0f);

    int lower = (int)ceilf(minp * Lf);
    int upper = (int)ceilf(maxp * Lf);

    // best_tz = max k<=E s.t. a multiple of 2^k lies in [lower, upper).
    // For lower>=1: exists mult of 2^k in (lower-1, upper-1] iff
    // (lower-1)>>k < (upper-1)>>k iff k <= msb((lower-1)^(upper-1)).
    float max_stab = 0.0f;
    if (lower < upper) {
        int tz;
        if (lower == 0) {
            tz = E;
        } else {
            unsigned x = (unsigned)((lower - 1) ^ (upper - 1)); // nonzero: A<B
            tz = 31 - __builtin_clz(x);
            tz = tz > E ? E : tz;
        }
        // 1 - 2^-tz, exact via exponent-field construction
        float p2 = __uint_as_float((unsigned)(127 - tz) << 23);
        max_stab = 1.0f - p2;
    }

    float stability = 1.0f - s * inv_sl; // inv_sl exact for pow2 stream_len
    return stability / max_stab;         // IEEE divide; /0 -> +inf like jnp
}

__global__ __launch_bounds__(256) void norm_stab_v4(
    const v4f* __restrict__ in_value, const v4f* __restrict__ stable_len,
    v4f* __restrict__ out, const int* __restrict__ stream_len_p, int nvec) {
    int i = blockIdx.x * blockDim.x + threadIdx.x;
    if (i >= nvec) return;

    int sl      = stream_len_p[0];                               // 1024
    int E       = (sl > 1) ? (32 - __builtin_clz((unsigned)(sl - 1))) : 0;
    float Lf    = (float)(1 << E);
    float inv_sl = 1.0f / (float)sl;                             // exact (pow2)

    v4f v = __builtin_nontemporal_load(&in_value[i]);   // global_load_b128 NT
    v4f s = __builtin_nontemporal_load(&stable_len[i]); // global_load_b128 NT

    v4f r;
#pragma unroll
    for (int j = 0; j < 4; ++j)
        r[j] = norm_stab_elem(v[j], s[j], E, Lf, inv_sl);

    __builtin_nontemporal_store(r, &out[i]);            // global_store_b128 NT
}

__global__ __launch_bounds__(64) void norm_stab_tail(
    const float* __restrict__ in_value, const float* __restrict__ stable_len,
    float* __restrict__ out, const int* __restrict__ stream_len_p,
    int first, int n) {
    int i = first + (int)threadIdx.x;
    if (i >= n) return;
    int sl       = stream_len_p[0];
    int E        = (sl > 1) ? (32 - __builtin_clz((unsigned)(sl - 1))) : 0;
    float Lf     = (float)(1 << E);
    float inv_sl = 1.0f / (float)sl;
    out[i] = norm_stab_elem(in_value[i], stable_len[i], E, Lf, inv_sl);
}

extern "C" void kernel_launch(void* const* d_in, const int* in_sizes, int n_in,
                              void* d_out, int out_size, void* d_ws, size_t ws_size,
                              hipStream_t stream) {
    (void)n_in; (void)d_ws; (void)ws_size; (void)out_size;
    const float* in_value   = (const float*)d_in[0];
    const float* stable_len = (const float*)d_in[1];
    const int*   stream_len = (const int*)d_in[2];
    float*       out        = (float*)d_out;

    int n    = in_sizes[0];       // 16*1024*1024
    int nvec = n >> 2;            // float4 groups
    int rem  = n & 3;

    if (nvec > 0) {
        int threads = 256;
        int blocks  = (nvec + threads - 1) / threads;
        norm_stab_v4<<<blocks, threads, 0, stream>>>(
            (const v4f*)in_value, (const v4f*)stable_len, (v4f*)out,
            stream_len, nvec);
    }
    if (rem > 0) {
        norm_stab_tail<<<1, 64, 0, stream>>>(in_value, stable_len, out,
                                             stream_len, nvec << 2, n);
    }
}